// TemporalXORNetwork_59717225283739
// MI455X (gfx1250) — compile-verified
//
#include <hip/hip_runtime.h>

typedef __attribute__((ext_vector_type(2))) float v2f;
typedef __attribute__((ext_vector_type(8))) float v8f;

#define T_STEPS 128
#define BATCH   8192
#define HID     256
#define BETA    0.9f
#define THRESH  1.0f
#define T_LAST  118   // last-10 window: t in [118, 127]

__global__ __launch_bounds__(512)
void snn_lif_wmma_kernel(const float* __restrict__ x,
                         const float* __restrict__ W1,
                         const float* __restrict__ W2,
                         float* __restrict__ out) {
    // LDS: x tile for this block's 16 batch rows, all 128 timesteps.
    // layout: xs[t*32 + bl*2 + i], 4096 floats = 16 KB
    __shared__ float xs[T_STEPS * 32];
    __shared__ float outAcc[32];   // [bl*2 + o]

    const int tid = threadIdx.x;
    const int b0  = blockIdx.x * 16;

    if (tid < 32) outAcc[tid] = 0.0f;

    // Cooperative, coalesced load of the x tile: each wave covers one
    // 128-byte row (one t) per iteration.
#pragma unroll
    for (int k = 0; k < 8; ++k) {
        int idx = tid + k * 512;
        int t   = idx >> 5;
        int rem = idx & 31;
        xs[idx] = x[(t * BATCH + b0) * 2 + rem];
    }
    __syncthreads();

    const int  lane = tid & 31;
    const int  wave = tid >> 5;      // 0..15 -> h tile
    const int  h0   = wave * 16;
    const int  l16  = lane & 15;
    const bool loHalf = lane < 16;
    const int  h    = h0 + l16;

    // B matrix (W1 tile), 4x16 f32: VGPR0 = {K0 | K2}, VGPR1 = {K1 | K3};
    // K=2,3 are zero padding (upper lanes).
    v2f bmat;
    bmat.x = loHalf ? W1[h * 2 + 0] : 0.0f;
    bmat.y = loHalf ? W1[h * 2 + 1] : 0.0f;

    // Output projection weights for this lane's h.
    const float w2_0 = W2[h];           // W2[0, h]
    const float w2_1 = W2[HID + h];     // W2[1, h]

    v8f mem = {0.f, 0.f, 0.f, 0.f, 0.f, 0.f, 0.f, 0.f};
    v8f cnt = {0.f, 0.f, 0.f, 0.f, 0.f, 0.f, 0.f, 0.f};

    const float* xrow = &xs[l16 * 2];

    for (int t = 0; t < T_STEPS; ++t) {
        // A matrix (x tile), 16x4 f32: lanes 0-15 hold K0/K1, lanes 16-31
        // hold K2/K3 (zero padding).
        v2f a;
        if (loHalf) {
            a.x = xrow[0];
            a.y = xrow[1];
        } else {
            a.x = 0.0f;
            a.y = 0.0f;
        }
        xrow += 32;

        // Fold leak + subtract-reset into the WMMA accumulator:
        //   C = beta*mem - (mem > TH)*TH ;  mem_new = x@W1 + C
        v8f c;
#pragma unroll
        for (int v = 0; v < 8; ++v) {
            float m = mem[v];
            c[v] = BETA * m - ((m > THRESH) ? THRESH : 0.0f);
        }

        mem = __builtin_amdgcn_wmma_f32_16x16x4_f32(
            /*neg_a=*/false, a, /*neg_b=*/false, bmat,
            /*c_mod=*/(short)0, c, /*reuse_a=*/false, /*reuse_b=*/false);

        if (t >= T_LAST) {
#pragma unroll
            for (int v = 0; v < 8; ++v)
                cnt[v] += (mem[v] > THRESH) ? 1.0f : 0.0f;
        }
    }

    // out[b, o] = sum_h (cnt/10) * W2[o, h]
    // C/D layout: reg v, lane L -> (b_local = v + 8*(L>=16), h = h0 + L%16).
    // Butterfly-reduce over the 16 lanes of each half, then accumulate
    // across waves via LDS float atomics (ds_add_f32).
#pragma unroll
    for (int v = 0; v < 8; ++v) {
        float avg = cnt[v] * 0.1f;
        float p0  = avg * w2_0;
        float p1  = avg * w2_1;
#pragma unroll
        for (int m = 1; m <= 8; m <<= 1) {
            p0 += __shfl_xor(p0, m, 32);
            p1 += __shfl_xor(p1, m, 32);
        }
        if (l16 == 0) {
            int bl = v + (loHalf ? 0 : 8);
            atomicAdd(&outAcc[bl * 2 + 0], p0);
            atomicAdd(&outAcc[bl * 2 + 1], p1);
        }
    }
    __syncthreads();

    if (tid < 32) out[b0 * 2 + tid] = outAcc[tid];
}

extern "C" void kernel_launch(void* const* d_in, const int* in_sizes, int n_in,
                              void* d_out, int out_size, void* d_ws, size_t ws_size,
                              hipStream_t stream) {
    (void)in_sizes; (void)n_in; (void)out_size; (void)d_ws; (void)ws_size;
    const float* x  = (const float*)d_in[0];   // [128, 8192, 2]
    const float* W1 = (const float*)d_in[1];   // [256, 2]
    const float* W2 = (const float*)d_in[2];   // [2, 256]
    float* out = (float*)d_out;                // [8192, 2]

    dim3 grid(BATCH / 16);   // 512 blocks, one 16-batch tile each
    dim3 block(512);         // 16 waves, one 16-h tile each
    snn_lif_wmma_kernel<<<grid, block, 0, stream>>>(x, W1, W2, out);
}